// SphericalConvolution_53979148976330
// MI455X (gfx1250) — compile-verified
//
#include <hip/hip_runtime.h>

// ---------------------------------------------------------------------------
// Spherical convolution: y[n,o,c] = (1/6)*sum_i x[n,i,c]*w[o,i,ord(c)] + bias[o]
//   x   : (8, 6*45, 16,16,16) f32   -> read exactly once  (35.4 MB)
//   w   : (48, 6, 5) f32            -> tiny, staged in LDS
//   bias: (48, 1) f32               -> tiny, staged in LDS
//   out : (8, 48*45, 16,16,16) f32  -> written exactly once (283 MB)
// Bandwidth-bound (~2.7 FLOP/B): stream with NT hints, do the K=6 contraction
// on the matrix pipe with V_WMMA_F32_16X16X4_F32 (K padded 6->8, zero weights).
// ---------------------------------------------------------------------------

typedef __attribute__((ext_vector_type(2))) float v2f;
typedef __attribute__((ext_vector_type(8))) float v8f;

#define SH_NCOEFF 45
#define SH_IN     6
#define SH_OUT    48
#define NORD      5
#define SPATIAL   4096                  // 16*16*16
#define CH_IN     (SH_IN * SH_NCOEFF)   // 270
#define CH_OUT    (SH_OUT * SH_NCOEFF)  // 2160
#define NTOT      (8 * SPATIAL)         // 32768 voxels
#define NTILES    (NTOT / 16)           // 2048 WMMA N-tiles
#define TPB       32                    // spatial tiles per block
#define WPB       8                     // waves per block (256 threads, wave32)
#define ITERS     (TPB / WPB)           // 4 tiles per wave

__global__ __launch_bounds__(256)
void sph_conv_wmma(const float* __restrict__ x,
                   const float* __restrict__ w,
                   const float* __restrict__ bias,
                   float* __restrict__ out) {
  __shared__ float lds_w[SH_OUT * SH_IN * NORD];  // 1440 floats
  __shared__ float lds_b[SH_OUT];

  const int tid = threadIdx.x;
  // Stage weight table + bias once per block (coalesced, hits L2 across blocks).
  for (int j = tid; j < SH_OUT * SH_IN * NORD; j += 256) lds_w[j] = w[j];
  if (tid < SH_OUT) lds_b[tid] = bias[tid];
  __syncthreads();

  const int c = blockIdx.x;  // coefficient 0..44
  // ord(c): band boundaries 1, 6, 15, 28 (hardcoded sh_indices)
  const int ord = (c >= 1) + (c >= 6) + (c >= 15) + (c >= 28);

  const int wave = tid >> 5;
  const int lane = tid & 31;
  const int half = lane >> 4;   // 0: lanes 0-15, 1: lanes 16-31
  const int lm   = lane & 15;

  // ---- A fragments (weights/6) and bias accumulators: built once per wave ----
  // A 16x4 f32 layout: lane = M (=output shell within tile);
  //   VGPR0 holds K = (half?2:0), VGPR1 holds K = (half?3:1).
  // WMMA #1 covers i=0..3; WMMA #2 covers i=4,5 with K=2,3 zero-padded in A.
  v2f a0[3], a1[3];
  v8f cb[3];
  const float inv6 = 1.0f / 6.0f;
#pragma unroll
  for (int t = 0; t < 3; ++t) {
    const int o = t * 16 + lm;
    const float* wo = &lds_w[o * (SH_IN * NORD) + ord];  // stride NORD per i
    const int i0 = half ? 2 : 0;
    a0[t].x = wo[(i0 + 0) * NORD] * inv6;
    a0[t].y = wo[(i0 + 1) * NORD] * inv6;
    a1[t].x = half ? 0.0f : wo[4 * NORD] * inv6;  // K=2,3 (i=6,7) -> 0
    a1[t].y = half ? 0.0f : wo[5 * NORD] * inv6;
    // D layout: VGPR r holds rows M = r (lanes 0-15) / r+8 (lanes 16-31)
#pragma unroll
    for (int r = 0; r < 8; ++r) cb[t][r] = lds_b[t * 16 + r + half * 8];
  }

  // ---- Stream 4 spatial tiles per wave; adjacent waves take adjacent tiles ----
#pragma unroll
  for (int it = 0; it < ITERS; ++it) {
    const int T  = blockIdx.y * TPB + it * WPB + wave;  // 0..2047
    const int n0 = T * 16;
    const int b  = n0 >> 12;      // batch (tiles never straddle a batch)
    const int s0 = n0 & 4095;     // spatial offset

    const float* xb = x + ((size_t)b * CH_IN + c) * SPATIAL + s0;

    // B fragments (K x N, N = 16 consecutive voxels), mirror-of-A layout:
    // VGPR0: rows K=0 | K=2 across lane halves; VGPR1: rows K=1 | K=3.
    v2f b0, b1;
    b0.x = __builtin_nontemporal_load(&xb[(size_t)(half ? 2 : 0) * SPATIAL + lm]);
    b0.y = __builtin_nontemporal_load(&xb[(size_t)(half ? 3 : 1) * SPATIAL + lm]);
    // Pad rows K=2,3 of 2nd WMMA replicate i=4,5 (finite * 0-weight = 0).
    b1.x = __builtin_nontemporal_load(&xb[(size_t)4 * SPATIAL + lm]);
    b1.y = __builtin_nontemporal_load(&xb[(size_t)5 * SPATIAL + lm]);

    float* ob = out + ((size_t)b * CH_OUT + c) * SPATIAL + s0;

#pragma unroll
    for (int t = 0; t < 3; ++t) {
      // D = A0*B0 + bias ; D = A1*B1 + D   (K = 6 split 4 + 2)
      v8f acc = __builtin_amdgcn_wmma_f32_16x16x4_f32(
          false, a0[t], false, b0, (short)0, cb[t], false, false);
      acc = __builtin_amdgcn_wmma_f32_16x16x4_f32(
          false, a1[t], false, b1, (short)0, acc, false, false);
      // Streaming stores: each store = two 64B contiguous segments (half-wave
      // per output row); 8 waves cover 512B contiguous spans per row.
#pragma unroll
      for (int r = 0; r < 8; ++r) {
        const int o = t * 16 + r + half * 8;
        __builtin_nontemporal_store(acc[r],
            &ob[(size_t)o * SH_NCOEFF * SPATIAL + lm]);
      }
    }
  }
}

extern "C" void kernel_launch(void* const* d_in, const int* in_sizes, int n_in,
                              void* d_out, int out_size, void* d_ws, size_t ws_size,
                              hipStream_t stream) {
  (void)in_sizes; (void)n_in; (void)out_size; (void)d_ws; (void)ws_size;
  const float* x    = (const float*)d_in[0];
  const float* w    = (const float*)d_in[1];
  const float* bias = (const float*)d_in[2];
  // d_in[3] = sh_indices (int64) — band structure is hardcoded in the kernel.
  float* out = (float*)d_out;

  dim3 grid(SH_NCOEFF, NTILES / TPB);  // (45, 64)
  sph_conv_wmma<<<grid, 256, 0, stream>>>(x, w, bias, out);
}